// FractionalDilationM2_18880676233730
// MI455X (gfx1250) — compile-verified
//
#include <hip/hip_runtime.h>
#include <math.h>
#include <stdint.h>

// Problem shape (fixed by reference setup_inputs)
#define B_    4
#define C_    32
#define OR_   8
#define H_    128
#define W_    128
#define HW_   (H_ * W_)
#define KVOL  125            // 5*5*5
#define KTOT  (C_ * OR_ * KVOL)   // 32000 floats in d_ws

// Tiling
#define TH_     32
#define TW_     32
#define LROW    36           // 32 + 2*2 halo
#define LPLANE  (LROW * LROW)      // 1296 floats
#define LDS_N   (OR_ * LPLANE)     // 10368 floats = 41472 B

// ---------------------------------------------------------------------------
// Kernel 1: build morphological structuring function k[c][o][a][j][i]
// k(g) = coeff * rho(g)^q  (see reference _morph_kernel)
// ---------------------------------------------------------------------------
__global__ __launch_bounds__(256) void build_morph_kernel(
    const float* __restrict__ mp, float* __restrict__ k) {
  int gid = blockIdx.x * 256 + threadIdx.x;
  if (gid >= KTOT) return;
  int c = gid / (OR_ * KVOL);
  int r = gid - c * (OR_ * KVOL);
  int o = r / KVOL;  r -= o * KVOL;
  int a = r / 25;    r -= a * 25;
  int j = r / 5;
  int i = r - j * 5;

  const double PI = 3.14159265358979323846;
  const double a2 = 1.3;                      // 2*ALPHA
  const float  q     = (float)(a2 / (a2 - 1.0));                       // 4.3333
  const float  coeff = (float)((a2 - 1.0) / a2 * pow(a2, -1.0 / (a2 - 1.0)));

  float dth = (float)(2.0 * PI / OR_) * (float)(a - 2);
  float dj  = (float)(j - 2);
  float di  = (float)(i - 2);
  float th  = (float)(2.0 * PI / OR_) * (float)o;
  float cth = cosf(th), sth = sinf(th);
  float xx  =  cth * di + sth * dj;
  float yy  = -sth * di + cth * dj;
  float half = 0.5f * dth;
  float cot  = (fabsf(half) < 1e-6f) ? 1.0f : half * cosf(half) / sinf(half);
  float c1 =  cot * xx + half * yy;
  float c2 = -half * xx + cot * yy;
  float c3 = dth;
  float t0 = mp[c * 3 + 0] * c1;
  float t1 = mp[c * 3 + 1] * c2;
  float t2 = mp[c * 3 + 2] * c3;
  float rho = sqrtf(t0 * t0 + t1 * t1 + t2 * t2 + 1e-12f);
  k[gid] = coeff * powf(rho, q);
}

// ---------------------------------------------------------------------------
// Kernel 2: M2 morphological dilation.
// One block = one (b,c) * 32x32 spatial tile * all 8 orientations.
// LDS staged via gfx1250 async global->LDS copies; -inf halo for spatial OOB.
// ---------------------------------------------------------------------------
__global__ __launch_bounds__(256) void m2_dilate(
    const float* __restrict__ x, const float* __restrict__ k,
    float* __restrict__ out) {
  __shared__ float tile[LDS_N];

  const int bid = blockIdx.x;
  const int tX = bid & 3;
  const int tY = (bid >> 2) & 3;
  const int c  = (bid >> 4) & 31;
  const int b  = bid >> 9;

  const float* __restrict__ xb = x + ((size_t)(b * C_ + c)) * (OR_ * HW_);
  const int h0 = tY * TH_ - 2;
  const int w0 = tX * TW_ - 2;
  const unsigned ldsb = (unsigned)(size_t)(void*)tile;  // LDS byte address = low 32 bits

  // ---- Stage tile (8 orientation planes, 36x36 each) into LDS ----
  for (int idx = threadIdx.x; idx < LDS_N; idx += 256) {
    int p   = idx / LPLANE;
    int rr  = idx - p * LPLANE;
    int row = rr / LROW;
    int col = rr - row * LROW;
    int h = h0 + row;
    int w = w0 + col;
    if ((unsigned)h < (unsigned)H_ && (unsigned)w < (unsigned)W_) {
      const float* g = xb + p * HW_ + h * W_ + w;
      // CDNA5 async copy: memory -> LDS, tracked by ASYNCcnt, no VGPR data.
      asm volatile("global_load_async_to_lds_b32 %0, %1, off"
                   :: "v"(ldsb + (unsigned)idx * 4u), "v"(g) : "memory");
    } else {
      tile[idx] = -INFINITY;   // spatial -inf padding
    }
  }
  asm volatile("s_wait_asynccnt 0" ::: "memory");
  __syncthreads();

  // ---- Compute: each thread -> 8 orientations x 4 consecutive W outputs ----
  const int tx   = threadIdx.x & 7;      // 0..7   -> W group
  const int ty   = threadIdx.x >> 3;     // 0..31  -> H row
  const int wloc = tx * 4;               // 0..28

  float acc[OR_][4];
#pragma unroll
  for (int o = 0; o < OR_; ++o)
#pragma unroll
    for (int w = 0; w < 4; ++w) acc[o][w] = -INFINITY;

  const float* __restrict__ kc = k + (size_t)c * (OR_ * KVOL);

#pragma unroll
  for (int p = 0; p < OR_; ++p) {          // input orientation plane
#pragma unroll 1
    for (int j = 0; j < 5; ++j) {          // kernel row (rolled: bounds code size)
      // 8-wide x row from LDS: covers i=0..4 taps for 4 outputs. 16B aligned.
      const float4* rp = (const float4*)(tile + p * LPLANE + (ty + j) * LROW + wloc);
      float4 xa = rp[0];
      float4 xc = rp[1];
      float xr[8] = {xa.x, xa.y, xa.z, xa.w, xc.x, xc.y, xc.z, xc.w};
#pragma unroll
      for (int o5 = 0; o5 < 5; ++o5) {
        const int o = (p + 6 + o5) & 7;    // output orientation p-2+o5 (mod 8)
        const int a = 4 - o5;              // kernel orientation offset
        // Wave-uniform address -> scalar (SMEM) loads.
        const float* __restrict__ kk = kc + o * KVOL + a * 25 + j * 5;
        float k0 = kk[0], k1 = kk[1], k2 = kk[2], k3 = kk[3], k4 = kk[4];
#pragma unroll
        for (int w = 0; w < 4; ++w) {
          float m = acc[o][w];
          m = fmaxf(m, fmaxf(xr[w + 0] - k0, xr[w + 1] - k1));  // -> v_max3_f32
          m = fmaxf(m, fmaxf(xr[w + 2] - k2, xr[w + 3] - k3));  // -> v_max3_f32
          m = fmaxf(m, xr[w + 4] - k4);
          acc[o][w] = m;
        }
      }
    }
  }

  // ---- Store: 8 coalesced float4 stores per thread ----
  const int hg = tY * TH_ + ty;
  const int wg = tX * TW_ + wloc;
#pragma unroll
  for (int o = 0; o < OR_; ++o) {
    float4 v;
    v.x = acc[o][0]; v.y = acc[o][1]; v.z = acc[o][2]; v.w = acc[o][3];
    *(float4*)(out + ((((size_t)(b * C_ + c)) * OR_ + o) * H_ + hg) * W_ + wg) = v;
  }
}

// ---------------------------------------------------------------------------
extern "C" void kernel_launch(void* const* d_in, const int* in_sizes, int n_in,
                              void* d_out, int out_size, void* d_ws, size_t ws_size,
                              hipStream_t stream) {
  const float* x  = (const float*)d_in[0];   // [4,32,8,128,128] f32
  const float* mp = (const float*)d_in[1];   // [32,3] f32
  float* out = (float*)d_out;                // [4,32,8,128,128] f32
  float* kbuf = (float*)d_ws;                // 32000 f32 = 128 KB scratch

  build_morph_kernel<<<(KTOT + 255) / 256, 256, 0, stream>>>(mp, kbuf);
  m2_dilate<<<B_ * C_ * 4 * 4, 256, 0, stream>>>(x, kbuf, out);
}